// GCN_36859409334384
// MI455X (gfx1250) — compile-verified
//
#include <hip/hip_runtime.h>

typedef float v2f __attribute__((ext_vector_type(2)));
typedef float v4f __attribute__((ext_vector_type(4)));
typedef float v8f __attribute__((ext_vector_type(8)));

#define D_IN  128
#define D_HID 128
#define D_OUT 64

// ---------------------------------------------------------------------------
// Dense GEMM + bias (+ optional fused ReLU on the A operand) using
// V_WMMA_F32_16X16X4_F32. One wave computes a 16-row strip across the full
// output width (NT tiles of 16 cols). Wave32; EXEC stays all-1s (only
// wave-uniform early exit).
//
// A-fragment layout (ISA 7.12.2, 32-bit A 16x4): lane<16 holds M=lane,K={0,1};
// lane>=16 holds M=lane-16,K={2,3} -> a single aligned float2 load per k-step.
// B-fragment mirrors with N striped across lanes. C/D: VGPR r, lanes 0-15 ->
// M=r, lanes 16-31 -> M=r+8, N = lane&15 (so bias is one scalar per lane).
// ---------------------------------------------------------------------------
template <int NT, bool RELU_A>
__global__ __launch_bounds__(128)
void gcn_gemm_wmma(const float* __restrict__ A, const float* __restrict__ B,
                   const float* __restrict__ bias, float* __restrict__ C,
                   int M, int K) {
  const int N     = NT * 16;
  const int lane  = threadIdx.x & 31;
  const int wave  = threadIdx.x >> 5;
  const int strip = blockIdx.x * 4 + wave;
  const int row0  = strip * 16;
  if (row0 >= M) return;  // wave-uniform

  const int half = lane >> 4;  // 0 or 1
  const int lm   = lane & 15;

  v8f acc[NT] = {};

  const float* arow = A + (size_t)(row0 + lm) * K;

  for (int k = 0; k < K; k += 4) {
    v2f av = *(const v2f*)(arow + k + 2 * half);
    if (RELU_A) {
      av.x = av.x > 0.f ? av.x : 0.f;
      av.y = av.y > 0.f ? av.y : 0.f;
    }
    const float* brow = B + (size_t)(k + 2 * half) * N + lm;
#pragma unroll
    for (int t = 0; t < NT; ++t) {
      v2f bv;
      bv.x = brow[t * 16];
      bv.y = brow[t * 16 + N];
      acc[t] = __builtin_amdgcn_wmma_f32_16x16x4_f32(
          /*neg_a=*/false, av, /*neg_b=*/false, bv,
          /*c_mod=*/(short)0, acc[t], /*reuse_a=*/false, /*reuse_b=*/false);
    }
  }

#pragma unroll
  for (int t = 0; t < NT; ++t) {
    const int   n     = t * 16 + lm;
    const float bs    = bias[n];
    float*      cbase = C + (size_t)(row0 + 8 * half) * N + n;
#pragma unroll
    for (int r = 0; r < 8; ++r) {
      cbase[(size_t)r * N] = acc[t][r] + bs;
    }
  }
}

// ---------------------------------------------------------------------------
// COO SpMM scatter: out[row] += val * H[col]. One wave per edge; each lane
// handles D/32 contiguous floats (float4 for D=128, float2 for D=64), so the
// gather is a fully coalesced 4*D-byte read and the scatter uses
// global_atomic_add_f32 into an L2-resident accumulator.
// ---------------------------------------------------------------------------
template <int D>
__global__ __launch_bounds__(256)
void gcn_spmm_atomic(const int* __restrict__ erow, const int* __restrict__ ecol,
                     const float* __restrict__ eval, const float* __restrict__ H,
                     float* __restrict__ Out, int nE) {
  const int e = blockIdx.x * (256 / 32) + (threadIdx.x >> 5);
  if (e >= nE) return;
  const int     lane = threadIdx.x & 31;
  constexpr int V    = D / 32;  // floats per lane

  const int   r = erow[e];
  const int   c = ecol[e];
  const float v = eval[e];

  const float* src = H + (size_t)c * D + lane * V;
  float*       dst = Out + (size_t)r * D + lane * V;

  if constexpr (V == 4) {
    v4f s = *(const v4f*)src;
    atomicAdd(dst + 0, v * s.x);
    atomicAdd(dst + 1, v * s.y);
    atomicAdd(dst + 2, v * s.z);
    atomicAdd(dst + 3, v * s.w);
  } else {
    v2f s = *(const v2f*)src;
    atomicAdd(dst + 0, v * s.x);
    atomicAdd(dst + 1, v * s.y);
  }
}

extern "C" void kernel_launch(void* const* d_in, const int* in_sizes, int n_in,
                              void* d_out, int out_size, void* d_ws, size_t ws_size,
                              hipStream_t stream) {
  const float* x    = (const float*)d_in[0];
  const float* W1   = (const float*)d_in[1];
  const float* b1   = (const float*)d_in[2];
  const float* W2   = (const float*)d_in[3];
  const float* b2   = (const float*)d_in[4];
  const int*   erow = (const int*)d_in[5];
  const int*   ecol = (const int*)d_in[6];
  const float* eval = (const float*)d_in[7];
  float*       out  = (float*)d_out;

  const int M = in_sizes[0] / D_IN;  // 100000 nodes
  const int E = in_sizes[5];         // 1600000 edges

  // Workspace layout: [h_pre: M*128][agg1: M*128]; h2 (M*64) reuses h_pre.
  float* h_pre = (float*)d_ws;
  float* agg1  = h_pre + (size_t)M * D_HID;
  float* h2    = h_pre;  // h_pre is dead after SpMM1

  hipMemsetAsync(agg1, 0, (size_t)M * D_HID * sizeof(float), stream);
  hipMemsetAsync(out, 0, (size_t)M * D_OUT * sizeof(float), stream);

  const int strips     = (M + 15) / 16;       // 6250
  const int gemmBlocks = (strips + 3) / 4;    // 4 waves/block
  const int spmmBlocks = (E + 7) / 8;         // 8 waves/block, 1 edge/wave

  // Layer 1: h_pre = x @ W1 + b1
  gcn_gemm_wmma<D_HID / 16, false><<<gemmBlocks, 128, 0, stream>>>(
      x, W1, b1, h_pre, M, D_IN);

  // agg1 = A_sparse @ h_pre
  gcn_spmm_atomic<D_HID><<<spmmBlocks, 256, 0, stream>>>(
      erow, ecol, eval, h_pre, agg1, E);

  // Layer 2: h2 = relu(agg1) @ W2 + b2  (ReLU fused into A-load)
  gcn_gemm_wmma<D_OUT / 16, true><<<gemmBlocks, 128, 0, stream>>>(
      agg1, W2, b2, h2, M, D_HID);

  // out = A_sparse @ h2
  gcn_spmm_atomic<D_OUT><<<spmmBlocks, 256, 0, stream>>>(
      erow, ecol, eval, h2, out, E);
}